// CompetitiveSelection_85504208929283
// MI455X (gfx1250) — compile-verified
//
#include <hip/hip_runtime.h>
#include <stdint.h>

#define COLS 2048
#define KSEL 256
#define NT   256
#define RPB  8   // rows per block

typedef int   v4i __attribute__((ext_vector_type(4)));
typedef float v4f __attribute__((ext_vector_type(4)));
typedef __attribute__((address_space(1))) v4i v4i_g;   // global int4
typedef __attribute__((address_space(3))) v4i v4i_l;   // LDS int4

#if defined(__AMDGCN__) && __has_builtin(__builtin_amdgcn_global_load_async_to_lds_b128)
#define HAVE_ASYNC 1
#else
#define HAVE_ASYNC 0
#endif

__device__ __forceinline__ void wait_async0() {
#if HAVE_ASYNC
#if __has_builtin(__builtin_amdgcn_s_wait_asynccnt)
  __builtin_amdgcn_s_wait_asynccnt(0);
#else
  asm volatile("s_wait_asynccnt 0x0" ::: "memory");
#endif
#endif
}

// Bring one 2048-float row into LDS: 2 x b128 per thread, 256 threads.
__device__ __forceinline__ void issue_row(const float* __restrict__ src,
                                          float* dst_lds, int tid) {
#if HAVE_ASYNC
  __builtin_amdgcn_global_load_async_to_lds_b128(
      (v4i_g*)(src + tid * 4), (v4i_l*)(dst_lds + tid * 4), 0, 0);
  __builtin_amdgcn_global_load_async_to_lds_b128(
      (v4i_g*)(src + 1024 + tid * 4), (v4i_l*)(dst_lds + 1024 + tid * 4), 0, 0);
#else
  ((v4f*)dst_lds)[tid]       = ((const v4f*)src)[tid];
  ((v4f*)dst_lds)[tid + 256] = ((const v4f*)src)[tid + 256];
  __builtin_prefetch(src + COLS, 0, 0);   // global_prefetch_b8 for next row
#endif
}

__global__ __launch_bounds__(NT) void competitive_select_kernel(
    const float* __restrict__ x, const float* __restrict__ imp,
    float* __restrict__ out, int total_rows) {

  __shared__ __align__(16) float s_x[2][COLS];   // double-buffered row tile
  __shared__ __align__(16) float s_imp[COLS];    // |importance|
  __shared__ unsigned s_hist[256];
  __shared__ unsigned s_sel[2];                  // [0]=selected bin, [1]=new remaining
  __shared__ unsigned s_cnt;

  const int tid  = threadIdx.x;
  const int row0 = blockIdx.x * RPB;
  if (row0 >= total_rows) return;

  // cache |importance| once per block
  for (int j = tid; j < COLS; j += NT) s_imp[j] = fabsf(imp[j]);

  // prefetch first row asynchronously into buffer 0
  issue_row(x + (size_t)row0 * COLS, s_x[0], tid);

  for (int r = 0; r < RPB; ++r) {
    const int row = row0 + r;
    if (row >= total_rows) break;
    const int cur = r & 1;

    wait_async0();        // per-wave: all outstanding async LDS fills done
    __syncthreads();      // all waves' fills visible; also fences prior row's LDS use

    // Pull my 8 elements into registers, compute score keys (monotonic uint).
    v4f xa = *(const v4f*)&s_x[cur][tid * 4];
    v4f xb = *(const v4f*)&s_x[cur][1024 + tid * 4];
    v4f ia = *(const v4f*)&s_imp[tid * 4];
    v4f ib = *(const v4f*)&s_imp[1024 + tid * 4];
    float xv[8] = {xa.x, xa.y, xa.z, xa.w, xb.x, xb.y, xb.z, xb.w};
    float iv[8] = {ia.x, ia.y, ia.z, ia.w, ib.x, ib.y, ib.z, ib.w};
    unsigned key[8];
#pragma unroll
    for (int e = 0; e < 8; ++e) key[e] = __float_as_uint(fabsf(xv[e]) * iv[e]);

    __syncthreads();      // everyone done reading s_x[cur] region before refills

    // Start fetching the next row into the other buffer (overlaps the select).
    if (r + 1 < RPB && row + 1 < total_rows)
      issue_row(x + (size_t)(row + 1) * COLS, s_x[cur ^ 1], tid);

    // ---- exact radix select of the KSEL-th largest key (4 x 8-bit MSB passes)
    unsigned prefix = 0, rem = KSEL;
#pragma unroll
    for (int shift = 24; shift >= 0; shift -= 8) {
      const unsigned decided = (shift == 24) ? 0u : (0xFFFFFFFFu << (shift + 8));
      s_hist[tid] = 0;
      __syncthreads();
#pragma unroll
      for (int e = 0; e < 8; ++e) {
        unsigned k = key[e];
        if ((k & decided) == prefix) atomicAdd(&s_hist[(k >> shift) & 255u], 1u);
      }
      __syncthreads();
      // Wave 0: suffix-scan the 256 bins with wave32 shuffles, pick threshold bin.
      if (tid < 32) {
        unsigned suf[8];
        unsigned acc = 0;
#pragma unroll
        for (int i = 7; i >= 0; --i) { acc += s_hist[tid * 8 + i]; suf[i] = acc; }
        unsigned total = acc;
        unsigned s = total;                  // inclusive suffix sum across lanes
#pragma unroll
        for (int d = 1; d < 32; d <<= 1) {
          unsigned o = __shfl_down(s, (unsigned)d, 32);
          if (tid + d < 32) s += o;
        }
        unsigned above = s - total;          // count in lanes (bins) above mine
#pragma unroll
        for (int i = 0; i < 8; ++i) {
          unsigned ge  = suf[i] + above;
          unsigned ge1 = (i == 7) ? above : (suf[i + 1] + above);
          if (ge >= rem && ge1 < rem) {      // exactly one (bin,lane) hits this
            s_sel[0] = (unsigned)(tid * 8 + i);
            s_sel[1] = rem - ge1;
          }
        }
      }
      __syncthreads();
      prefix |= s_sel[0] << shift;
      rem = s_sel[1];
    }
    const unsigned T = prefix;               // exact K-th largest key
    const unsigned need = rem;               // how many ties at T to keep

    if (tid == 0) s_cnt = 0;
    __syncthreads();

    float o[8];
#pragma unroll
    for (int e = 0; e < 8; ++e) {
      unsigned k = key[e];
      float v = 0.0f;
      if (k > T) v = xv[e];
      else if (k == T) { if (atomicAdd(&s_cnt, 1u) < need) v = xv[e]; }
      o[e] = v;
    }
    float* orow = out + (size_t)row * COLS;
    v4f o0 = {o[0], o[1], o[2], o[3]};
    v4f o1 = {o[4], o[5], o[6], o[7]};
    __builtin_nontemporal_store(o0, (v4f*)(orow + tid * 4));
    __builtin_nontemporal_store(o1, (v4f*)(orow + 1024 + tid * 4));
  }
}

extern "C" void kernel_launch(void* const* d_in, const int* in_sizes, int n_in,
                              void* d_out, int out_size, void* d_ws, size_t ws_size,
                              hipStream_t stream) {
  const float* x   = (const float*)d_in[0];
  const float* imp = (const float*)d_in[1];
  float* out = (float*)d_out;
  const int rows = in_sizes[0] / COLS;               // 32768
  const int blocks = (rows + RPB - 1) / RPB;         // 4096
  hipLaunchKernelGGL(competitive_select_kernel, dim3(blocks), dim3(NT), 0, stream,
                     x, imp, out, rows);
}